// GaussianMixture_5549097746714
// MI455X (gfx1250) — compile-verified
//
#include <hip/hip_runtime.h>
#include <math.h>

typedef __attribute__((ext_vector_type(2))) float v2f;
typedef __attribute__((ext_vector_type(8))) float v8f;
typedef unsigned int v4u __attribute__((ext_vector_type(4)));
typedef int v4i __attribute__((ext_vector_type(4)));
typedef int v8i __attribute__((ext_vector_type(8)));

#define GMM_N 524288
#define GMM_K 32
#define GMM_D 64
#define LOG2PI 1.8378770664093453f

// workspace layout (floats)
#define WS_W    0      // 32 x 128: f<64 -> -0.5/var ; f>=64 -> mu/var
#define WS_CST  4096   // 32: log(pi_k) - 0.5*(logdet_k + mu'inv_var mu)
#define WS_RED  4128   // 4128 reduced: [rx 2048][rx2 2048][dem 32]
#define WS_PART 8256   // NB * RECLEN per-block partials
#define RECLEN  4128

#if defined(__has_builtin)
#if __has_builtin(__builtin_amdgcn_tensor_load_to_lds) && __has_builtin(__builtin_amdgcn_s_wait_tensorcnt)
#define USE_TDM 1
#endif
#endif
#ifndef USE_TDM
#define USE_TDM 0
#endif

__device__ __forceinline__ v8f wmma4(v2f a, v2f b, v8f c) {
  // V_WMMA_F32_16X16X4_F32 : D(16x16 f32) = A(16x4 f32) * B(4x16 f32) + C
  return __builtin_amdgcn_wmma_f32_16x16x4_f32(false, a, false, b, (short)0, c,
                                               false, false);
}

#if USE_TDM
// DMA one 64x64 fp32 tile (row-major, stride 64) into LDS at lds_off with
// 4-DWORD padding after every 64 DWORDs (=> LDS row stride 68 floats).
__device__ __forceinline__ void tdm_load_tile(const float* gsrc, unsigned lds_off) {
  unsigned long long ga = (unsigned long long)(uintptr_t)gsrc;
  v4u g0;
  g0.x = 1u;                                   // count=1, user mode
  g0.y = lds_off;                              // D#.lds_addr (bytes)
  g0.z = (unsigned)(ga & 0xFFFFFFFFu);         // global_addr[31:0]
  g0.w = (unsigned)((ga >> 32) & 0x1FFFFFFu)   // global_addr[56:32]
       | (2u << 30);                           // type = 2 ("image")
  v8i g1;
  g1[0] = (int)((2u << 16)      // data_size = 4 bytes
              | (1u << 20)      // pad_enable
              | (5u << 22)      // pad_interval: 64 DWORDs
              | (3u << 25));    // pad_amount : 4 DWORDs
  g1[1] = (int)(64u << 16);                       // tensor_dim0 = 64 (lo16)
  g1[2] = (int)(((unsigned)GMM_N & 0xFFFFu) << 16); // tensor_dim1 lo16
  g1[3] = (int)(((unsigned)GMM_N >> 16) | (64u << 16)); // tensor_dim1 hi, tile_dim0=64
  g1[4] = 64;                                     // tile_dim1 = 64, tile_dim2 = 0
  g1[5] = 64;                                     // tensor_dim0_stride = 64
  g1[6] = 0;
  g1[7] = 0;
  v4i z4 = {0, 0, 0, 0};
#if __clang_major__ >= 23
  v8i z8 = {0, 0, 0, 0, 0, 0, 0, 0};
  __builtin_amdgcn_tensor_load_to_lds(g0, g1, z4, z4, z8, 0);
#else
  __builtin_amdgcn_tensor_load_to_lds(g0, g1, z4, z4, 0);
#endif
}
#endif

__global__ void gmm_prep(const float* __restrict__ mu, const float* __restrict__ var,
                         const float* __restrict__ pi, float* __restrict__ ws) {
  int t = threadIdx.x;
  for (int idx = t; idx < GMM_K * 128; idx += 256) {
    int k = idx >> 7, f = idx & 127, d = f & 63;
    float iv = 1.0f / var[k * 64 + d];
    ws[WS_W + idx] = (f < 64) ? (-0.5f * iv) : (mu[k * 64 + d] * iv);
  }
  if (t < GMM_K) {
    float ld = 0.f, c1 = 0.f;
    for (int d = 0; d < 64; ++d) {
      float v = var[t * 64 + d];
      float m = mu[t * 64 + d];
      ld += logf(v) + LOG2PI;
      c1 += m * m / v;
    }
    ws[WS_CST + t] = logf(pi[t]) - 0.5f * (ld + c1);
  }
}

__global__ __launch_bounds__(128) void gmm_main(const float* __restrict__ X,
                                                float* __restrict__ ws,
                                                int numTiles) {
  __shared__ float sX[2][64 * 68]; // [n][d], stride 68, double buffered for TDM
  __shared__ float sW[32 * 130];   // [k][f], stride 130
  __shared__ float sR[32 * 66];    // [k][n], stride 66
  __shared__ float sDem[128];

  const int t = threadIdx.x;
  const int lane = t & 31;
  const int wave = t >> 5;
  const int lh = lane >> 4;   // lane half (0/1)
  const int ln = lane & 15;

  for (int idx = t; idx < 32 * 128; idx += 128) {
    int k = idx >> 7, f = idx & 127;
    sW[k * 130 + f] = ws[WS_W + idx];
  }
  v8f cst0, cst1;
#pragma unroll
  for (int j = 0; j < 8; ++j) {
    cst0[j] = ws[WS_CST + j + 8 * lh];
    cst1[j] = ws[WS_CST + 16 + j + 8 * lh];
  }

  v8f aRX0 = {0.f,0.f,0.f,0.f,0.f,0.f,0.f,0.f};
  v8f aRX1 = aRX0, aRX20 = aRX0, aRX21 = aRX0;
  float demAcc = 0.f;
  int buf = 0;

#if USE_TDM
  const unsigned ldsBase0 = (unsigned)(uintptr_t)&sX[0][0];
  const unsigned ldsBase1 = (unsigned)(uintptr_t)&sX[1][0];
  if (wave == 0 && blockIdx.x < numTiles)
    tdm_load_tile(X + (size_t)blockIdx.x * (64 * 64), ldsBase0);
#endif
  __syncthreads();

  for (int tile = blockIdx.x; tile < numTiles; tile += gridDim.x) {
    const int ntile = tile + gridDim.x;
#if USE_TDM
    if (wave == 0) {
      if (ntile < numTiles) {
        tdm_load_tile(X + (size_t)ntile * (64 * 64), buf ? ldsBase0 : ldsBase1);
        __builtin_amdgcn_s_wait_tensorcnt(1);  // current tile's DMA complete
      } else {
        __builtin_amdgcn_s_wait_tensorcnt(0);
      }
    }
#else
    {
      const float* src = X + (size_t)tile * (64 * 64);
      if (ntile < numTiles)
        __builtin_prefetch(X + (size_t)ntile * (64 * 64) + t * 32, 0, 0);
#pragma unroll
      for (int i = 0; i < 8; ++i) {
        int f = t + i * 128;
        float4 v = reinterpret_cast<const float4*>(src)[f];
        int n = f >> 4, c = (f & 15) * 4;
        *reinterpret_cast<float4*>(&sX[0][n * 68 + c]) = v;
      }
    }
#endif
    __syncthreads();
    const float* xb = sX[buf];

    // ---- E-step: log_post(32 x 16) = cst + W(32x128) * Y(128x16), per wave
    const int nloc = wave * 16 + ln;
    const float* xrow = &xb[nloc * 68];
    v8f acc0 = cst0, acc1 = cst1;
#pragma unroll 4
    for (int d0 = 0; d0 < 64; d0 += 4) {       // x^2 features (f = d0)
      v2f b, a0, a1;
      float x0 = xrow[d0 + 2 * lh];
      float x1 = xrow[d0 + 2 * lh + 1];
      b.x = x0 * x0; b.y = x1 * x1;
      a0.x = sW[ln * 130 + d0 + 2 * lh];        a0.y = sW[ln * 130 + d0 + 2 * lh + 1];
      a1.x = sW[(ln + 16) * 130 + d0 + 2 * lh]; a1.y = sW[(ln + 16) * 130 + d0 + 2 * lh + 1];
      acc0 = wmma4(a0, b, acc0);
      acc1 = wmma4(a1, b, acc1);
    }
#pragma unroll 4
    for (int d0 = 0; d0 < 64; d0 += 4) {       // x features (f = 64 + d0)
      int f0 = 64 + d0;
      v2f b, a0, a1;
      b.x = xrow[d0 + 2 * lh];
      b.y = xrow[d0 + 2 * lh + 1];
      a0.x = sW[ln * 130 + f0 + 2 * lh];        a0.y = sW[ln * 130 + f0 + 2 * lh + 1];
      a1.x = sW[(ln + 16) * 130 + f0 + 2 * lh]; a1.y = sW[(ln + 16) * 130 + f0 + 2 * lh + 1];
      acc0 = wmma4(a0, b, acc0);
      acc1 = wmma4(a1, b, acc1);
    }

    // ---- logsumexp over K=32: 16 k's local, 16 in partner lane (xor 16)
    float mx = acc0[0];
#pragma unroll
    for (int j = 0; j < 8; ++j) { mx = fmaxf(mx, acc0[j]); mx = fmaxf(mx, acc1[j]); }
    mx = fmaxf(mx, __shfl_xor(mx, 16, 32));
    float s = 0.f;
#pragma unroll
    for (int j = 0; j < 8; ++j) {
      acc0[j] = __expf(acc0[j] - mx); s += acc0[j];
      acc1[j] = __expf(acc1[j] - mx); s += acc1[j];
    }
    s += __shfl_xor(s, 16, 32);
    float inv = 1.0f / s;
#pragma unroll
    for (int j = 0; j < 8; ++j) {
      int k0 = j + 8 * lh;
      sR[k0 * 66 + nloc]        = acc0[j] * inv;
      sR[(16 + k0) * 66 + nloc] = acc1[j] * inv;
    }
    __syncthreads();

    // dem partial: thread t owns k = t>>2, n-slice (t&3)*16..+16
    {
      int k = t >> 2, nb = (t & 3) * 16;
      float dsum = 0.f;
#pragma unroll
      for (int i = 0; i < 16; ++i) dsum += sR[k * 66 + nb + i];
      demAcc += dsum;
    }

    // ---- M-step: rx/rx2 (32 x 16-d-slice per wave) += r(32x64) * X(64x16)
    const int dcol = wave * 16 + ln;
#pragma unroll 4
    for (int n0 = 0; n0 < 64; n0 += 4) {
      v2f b, bs, a0, a1;
      b.x = xb[(n0 + 2 * lh) * 68 + dcol];
      b.y = xb[(n0 + 2 * lh + 1) * 68 + dcol];
      bs.x = b.x * b.x; bs.y = b.y * b.y;
      a0.x = sR[ln * 66 + n0 + 2 * lh];         a0.y = sR[ln * 66 + n0 + 2 * lh + 1];
      a1.x = sR[(ln + 16) * 66 + n0 + 2 * lh];  a1.y = sR[(ln + 16) * 66 + n0 + 2 * lh + 1];
      aRX0  = wmma4(a0, b,  aRX0);
      aRX1  = wmma4(a1, b,  aRX1);
      aRX20 = wmma4(a0, bs, aRX20);
      aRX21 = wmma4(a1, bs, aRX21);
    }
    __syncthreads();   // all waves done with sX[buf] and sR
#if USE_TDM
    buf ^= 1;
#endif
  }

  // per-block partial record: [rx 32x64][rx2 32x64][dem 32]
  float* rec = ws + WS_PART + (size_t)blockIdx.x * RECLEN;
  const int d = wave * 16 + ln;
#pragma unroll
  for (int j = 0; j < 8; ++j) {
    int k0 = j + 8 * lh;
    rec[k0 * 64 + d]               = aRX0[j];
    rec[(16 + k0) * 64 + d]        = aRX1[j];
    rec[2048 + k0 * 64 + d]        = aRX20[j];
    rec[2048 + (16 + k0) * 64 + d] = aRX21[j];
  }
  sDem[t] = demAcc;
  __syncthreads();
  if (t < 32)
    rec[4096 + t] = sDem[t * 4] + sDem[t * 4 + 1] + sDem[t * 4 + 2] + sDem[t * 4 + 3];
}

__global__ void gmm_reduce(float* __restrict__ ws, int nb) {
  int e = blockIdx.x * blockDim.x + threadIdx.x;
  if (e >= RECLEN) return;
  float s = 0.f;
  const float* p = ws + WS_PART + e;
  for (int b = 0; b < nb; ++b) s += p[(size_t)b * RECLEN];
  ws[WS_RED + e] = s;
}

__global__ void gmm_final(const float* __restrict__ ws, float* __restrict__ out) {
  const float* red = ws + WS_RED;
  int t = threadIdx.x;
  if (t < 32) out[t * 129] = red[4096 + t] * (1.0f / (float)GMM_N);
  for (int idx = t; idx < GMM_K * GMM_D; idx += 256) {
    int k = idx >> 6, d = idx & 63;
    float dem = red[4096 + k];
    float rx  = red[idx];
    float rx2 = red[2048 + idx];
    float m = rx / dem;
    out[k * 129 + 1 + d]  = m;
    out[k * 129 + 65 + d] = rx2 / dem - m * m;
  }
}

extern "C" void kernel_launch(void* const* d_in, const int* in_sizes, int n_in,
                              void* d_out, int out_size, void* d_ws, size_t ws_size,
                              hipStream_t stream) {
  const float* X   = (const float*)d_in[0];
  const float* mu  = (const float*)d_in[1];
  const float* var = (const float*)d_in[2];
  const float* pi  = (const float*)d_in[3];
  float* ws  = (float*)d_ws;
  float* out = (float*)d_out;

  int n = in_sizes[0] / GMM_D;
  int numTiles = n / 64;

  long cap = (long)(ws_size / sizeof(float)) - WS_PART;
  int nb = (int)(cap / RECLEN);
  if (nb < 1) nb = 1;
  if (nb > 512) nb = 512;
  if (nb > numTiles) nb = numTiles;

  gmm_prep<<<1, 256, 0, stream>>>(mu, var, pi, ws);
  gmm_main<<<nb, 128, 0, stream>>>(X, ws, numTiles);
  gmm_reduce<<<(RECLEN + 255) / 256, 256, 0, stream>>>(ws, nb);
  gmm_final<<<1, 256, 0, stream>>>(ws, out);
}